// SAGESparseLayer_54863912239174
// MI455X (gfx1250) — compile-verified
//
#include <hip/hip_runtime.h>

#define N_NODES 100000
#define N_EDGES 625000
#define DFEAT   128
#define KDIM    256        // 2*DFEAT
#define ODIM    128
#define LDSTRIDE 262       // padded K-stride (even -> v2f/b64 aligned, 262%64=6 spreads banks)

typedef float v2f __attribute__((ext_vector_type(2)));
typedef float v8f __attribute__((ext_vector_type(8)));

// ---------------------------------------------------------------------------
// Phase 0: zero segment-sum + count scratch
// ---------------------------------------------------------------------------
__global__ void zero_ws_kernel(float* __restrict__ p, int n) {
    int i = blockIdx.x * blockDim.x + threadIdx.x;
    if (i < n) p[i] = 0.0f;
}

// ---------------------------------------------------------------------------
// Phase 1: edge scatter. One wave32 per edge; each lane handles 4 floats.
// seg_sum[dst] += feature[src] * w ; cnt[dst] += 1
// Working set (~160MB) is L2-resident on MI455X (192MB L2), so these atomics
// are L2 ops, not HBM round-trips.
// ---------------------------------------------------------------------------
__global__ __launch_bounds__(256) void scatter_edges_kernel(
    const float* __restrict__ feat,
    const float* __restrict__ ew,
    const int*   __restrict__ dst,
    const int*   __restrict__ src,
    float* __restrict__ segsum,
    float* __restrict__ cnt) {
    int lane = threadIdx.x & 31;
    int wave = threadIdx.x >> 5;
    int e = blockIdx.x * 8 + wave;           // 625000 = 78125 * 8, exact
    if (e >= N_EDGES) return;
    int s  = src[e];
    int d  = dst[e];
    float w = ew[e];
    const float4* frow = (const float4*)(feat + (size_t)s * DFEAT);
    float4 v = frow[lane];                   // lanes 0..31 cover 128 floats
    float* o = segsum + (size_t)d * DFEAT + lane * 4;
    atomicAdd(o + 0, v.x * w);
    atomicAdd(o + 1, v.y * w);
    atomicAdd(o + 2, v.z * w);
    atomicAdd(o + 3, v.w * w);
    if (lane == 0) atomicAdd(cnt + d, 1.0f);
}

// ---------------------------------------------------------------------------
// Phase 2: out = [feature | segsum/max(cnt,1)] @ W + b  via v_wmma_f32_16x16x4
// Block = 256 threads (8 waves). Block tile: 256 rows x 128 cols.
// Wave tile: 32 rows (two 16-row strips) x 128 cols = 16 accumulators;
// each LDS B-fragment feeds TWO WMMAs. Per k-step, all 8 B-fragments are
// loaded into registers as one DS clause before the 16 WMMAs, so dscnt waits
// are descending (0x7..0x0) and LDS latency hides behind in-flight WMMAs.
// W (256x128) staged transposed in LDS via async global->LDS scatter copies.
// ---------------------------------------------------------------------------
__global__ __launch_bounds__(256) void sage_gemm_kernel(
    const float* __restrict__ feat,
    const float* __restrict__ segsum,
    const float* __restrict__ cnt,
    const float* __restrict__ W,      // [256][128] row-major
    const float* __restrict__ bias,   // [128]
    float* __restrict__ out) {        // [N][128]
    extern __shared__ float Wlds[];   // ODIM * LDSTRIDE floats

    // ---- Stage W transposed (Wlds[n*LDSTRIDE + k] = W[k*128 + n]) using the
    // gfx1250 async global->LDS path: per-lane global src AND per-lane LDS dst,
    // tracked by ASYNCcnt. Low 32 bits of a generic LDS pointer == LDS offset.
    {
        unsigned lbase = (unsigned)(uintptr_t)(void*)Wlds;
        for (int i = threadIdx.x; i < KDIM * ODIM; i += 256) {
            int k = i >> 7;        // 0..255
            int n = i & 127;       // 0..127
            unsigned loff = lbase + (unsigned)((n * LDSTRIDE + k) * 4);
            unsigned long long ga = (unsigned long long)(uintptr_t)(W + i);
            asm volatile("global_load_async_to_lds_b32 %0, %1, off"
                         :: "v"(loff), "v"(ga) : "memory");
        }
        asm volatile("s_wait_asynccnt 0x0" ::: "memory");
    }
    __syncthreads();

    const int lane = threadIdx.x & 31;
    const int wave = threadIdx.x >> 5;
    const int half = lane >> 4;        // 0: K pair {k,k+1}; 1: {k+2,k+3}
    const int m    = lane & 15;        // row within strip / col within tile
    const int rowT0 = blockIdx.x * 256 + wave * 32;   // strip 0
    const int rowT1 = rowT0 + 16;                     // strip 1

    int r0 = rowT0 + m; int r0c = r0 < N_NODES ? r0 : (N_NODES - 1);
    int r1 = rowT1 + m; int r1c = r1 < N_NODES ? r1 : (N_NODES - 1);

    const float invc0 = 1.0f / fmaxf(cnt[r0c], 1.0f);
    const float invc1 = 1.0f / fmaxf(cnt[r1c], 1.0f);
    const float* f0 = feat   + (size_t)r0c * DFEAT;
    const float* f1 = feat   + (size_t)r1c * DFEAT;
    const float* s0 = segsum + (size_t)r0c * DFEAT;
    const float* s1 = segsum + (size_t)r1c * DFEAT;

    v8f acc0[8] = {};
    v8f acc1[8] = {};

    // ---- K = 0..127 : self features ----
    for (int k = 0; k < DFEAT; k += 4) {
        int kk = k + 2 * half;
        v2f a0 = *(const v2f*)(f0 + kk);
        v2f a1 = *(const v2f*)(f1 + kk);
        v2f bfr[8];
#pragma unroll
        for (int t = 0; t < 8; ++t)
            bfr[t] = *(const v2f*)(&Wlds[(t * 16 + m) * LDSTRIDE + kk]);
#pragma unroll
        for (int t = 0; t < 8; ++t) {
            acc0[t] = __builtin_amdgcn_wmma_f32_16x16x4_f32(
                false, a0, false, bfr[t], (short)0, acc0[t], false, false);
            acc1[t] = __builtin_amdgcn_wmma_f32_16x16x4_f32(
                false, a1, false, bfr[t], (short)0, acc1[t], false, false);
        }
    }
    // ---- K = 128..255 : neighbor mean (normalized on the fly) ----
    for (int k = 0; k < DFEAT; k += 4) {
        int kk = k + 2 * half;
        v2f q0 = *(const v2f*)(s0 + kk);
        v2f q1 = *(const v2f*)(s1 + kk);
        v2f a0; a0.x = q0.x * invc0; a0.y = q0.y * invc0;
        v2f a1; a1.x = q1.x * invc1; a1.y = q1.y * invc1;
        int kb = DFEAT + kk;
        v2f bfr[8];
#pragma unroll
        for (int t = 0; t < 8; ++t)
            bfr[t] = *(const v2f*)(&Wlds[(t * 16 + m) * LDSTRIDE + kb]);
#pragma unroll
        for (int t = 0; t < 8; ++t) {
            acc0[t] = __builtin_amdgcn_wmma_f32_16x16x4_f32(
                false, a0, false, bfr[t], (short)0, acc0[t], false, false);
            acc1[t] = __builtin_amdgcn_wmma_f32_16x16x4_f32(
                false, a1, false, bfr[t], (short)0, acc1[t], false, false);
        }
    }

    // ---- Store C tiles: VGPR v -> row (v + 8*half), col (lane&15), + bias ----
#pragma unroll
    for (int t = 0; t < 8; ++t) {
        int n  = t * 16 + m;
        float bv = bias[n];
#pragma unroll
        for (int v = 0; v < 8; ++v) {
            int ra = rowT0 + v + 8 * half;
            int rb = rowT1 + v + 8 * half;
            if (ra < N_NODES) out[(size_t)ra * ODIM + n] = acc0[t][v] + bv;
            if (rb < N_NODES) out[(size_t)rb * ODIM + n] = acc1[t][v] + bv;
        }
    }
}

// ---------------------------------------------------------------------------
extern "C" void kernel_launch(void* const* d_in, const int* in_sizes, int n_in,
                              void* d_out, int out_size, void* d_ws, size_t ws_size,
                              hipStream_t stream) {
    const float* feat = (const float*)d_in[0];   // [100000,128]
    const float* ew   = (const float*)d_in[1];   // [625000,1]
    const float* W    = (const float*)d_in[2];   // [256,128]
    const float* bias = (const float*)d_in[3];   // [128]
    const int*   ridx = (const int*)d_in[4];     // [2,625000] (dst row 0, src row 1)

    float* segsum = (float*)d_ws;                         // [N,128]
    float* cnt    = segsum + (size_t)N_NODES * DFEAT;     // [N]

    int totalz = N_NODES * DFEAT + N_NODES;
    zero_ws_kernel<<<(totalz + 255) / 256, 256, 0, stream>>>((float*)d_ws, totalz);

    scatter_edges_kernel<<<N_EDGES / 8, 256, 0, stream>>>(
        feat, ew, ridx, ridx + N_EDGES, segsum, cnt);

    size_t lds_bytes = (size_t)ODIM * LDSTRIDE * sizeof(float);  // ~131 KB
    sage_gemm_kernel<<<(N_NODES + 255) / 256, 256, lds_bytes, stream>>>(
        feat, segsum, cnt, W, bias, (float*)d_out);
}